// PatchCore_27041114096326
// MI455X (gfx1250) — compile-verified
//
#include <hip/hip_runtime.h>
#include <hip/hip_bf16.h>
#include <math.h>

typedef float v2f __attribute__((ext_vector_type(2)));
typedef float v8f __attribute__((ext_vector_type(8)));

#define DIMS 384
#define OUT_HW 224
#define IN_HW 28
#define KNN_K 9
#define GRAD 16   // gaussian radius: int(4*4.0+0.5)=16, 33 taps

// ---------------------------------------------------------------------------
// Kernel 1: row squared-norms  (one wave per row, 12 coalesced loads/lane)
// ---------------------------------------------------------------------------
__global__ __launch_bounds__(256) void pc_row_norm(const float* __restrict__ A,
                                                   float* __restrict__ out, int rows) {
  int wid  = (blockIdx.x * blockDim.x + threadIdx.x) >> 5;
  int lane = threadIdx.x & 31;
  if (wid >= rows) return;
  const float* r = A + (size_t)wid * DIMS;
  float s = 0.f;
  #pragma unroll
  for (int k = lane; k < DIMS; k += 32) { float v = r[k]; s += v * v; }
  #pragma unroll
  for (int o = 16; o > 0; o >>= 1) s += __shfl_xor(s, o, 32);
  if (lane == 0) out[wid] = s;
}

// ---------------------------------------------------------------------------
// Kernel 2: fused GEMM (WMMA f32 16x16x4) + min/argmin of (y2 - 2*x.y)
// One block = 32 query rows (two 16x16 A tiles -> each B load feeds 2 WMMAs).
// 8 waves split the bank columns in groups of 64 (four 16-wide N tiles).
// LDS: 48KB query tile, aliased by the 32KB min/argmin reduction buffer.
// ---------------------------------------------------------------------------
__global__ __launch_bounds__(256) void pc_knn_wmma(const float* __restrict__ X,
                                                   const float* __restrict__ Y,
                                                   const float* __restrict__ x2,
                                                   const float* __restrict__ y2,
                                                   float* __restrict__ pscore,
                                                   int* __restrict__ ploc,
                                                   int Q, int N) {
  __shared__ __align__(16) float smem[32 * DIMS];   // 48 KB (tile, then reduce)

  const int tid = threadIdx.x;
  const int qm  = blockIdx.x * 32;

  for (int i = tid; i < 32 * DIMS; i += 256) {
    int r = i / DIMS, c = i - r * DIMS;
    int gq = qm + r; if (gq > Q - 1) gq = Q - 1;
    smem[i] = X[(size_t)gq * DIMS + c];
  }
  __syncthreads();

  const int wave = tid >> 5;
  const int lane = tid & 31;
  const int hi   = lane >> 4;       // K half (A/B), M half (C/D)
  const int lr   = lane & 15;       // M row (A) / N col (B,C,D)

  float minv0[8], minv1[8]; int mini0[8], mini1[8];
  #pragma unroll
  for (int r = 0; r < 8; ++r) {
    minv0[r] = INFINITY; mini0[r] = 0;
    minv1[r] = INFINITY; mini1[r] = 0;
  }

  const float* xrow0 = &smem[lr * DIMS + 2 * hi];          // M-tile 0 (rows 0..15)
  const float* xrow1 = &smem[(lr + 16) * DIMS + 2 * hi];   // M-tile 1 (rows 16..31)
  const int ngroups = (N + 63) >> 6;

  for (int g = wave; g < ngroups; g += 8) {
    const int nb = g * 64;
    const int c0 = nb + lr, c1 = nb + 16 + lr, c2 = nb + 32 + lr, c3 = nb + 48 + lr;
    // clamp bases so OOB columns read valid memory; masked via y2=+inf below
    const float* yb0 = Y + (size_t)(c0 < N ? c0 : N - 1) * DIMS + 2 * hi;
    const float* yb1 = Y + (size_t)(c1 < N ? c1 : N - 1) * DIMS + 2 * hi;
    const float* yb2 = Y + (size_t)(c2 < N ? c2 : N - 1) * DIMS + 2 * hi;
    const float* yb3 = Y + (size_t)(c3 < N ? c3 : N - 1) * DIMS + 2 * hi;

    v8f a0c0 = {0,0,0,0,0,0,0,0}, a0c1 = {0,0,0,0,0,0,0,0};
    v8f a0c2 = {0,0,0,0,0,0,0,0}, a0c3 = {0,0,0,0,0,0,0,0};
    v8f a1c0 = {0,0,0,0,0,0,0,0}, a1c1 = {0,0,0,0,0,0,0,0};
    v8f a1c2 = {0,0,0,0,0,0,0,0}, a1c3 = {0,0,0,0,0,0,0,0};

    #pragma unroll 4
    for (int k = 0; k < DIMS; k += 4) {
      v2f a0 = *(const v2f*)(xrow0 + k);  // A: (M=lr,    K=k+2*hi .. +1)
      v2f a1 = *(const v2f*)(xrow1 + k);  // A: (M=lr+16, K=k+2*hi .. +1)
      v2f b0 = *(const v2f*)(yb0 + k);    // B: (K=k+2*hi .. +1, N=lr)
      v2f b1 = *(const v2f*)(yb1 + k);
      v2f b2 = *(const v2f*)(yb2 + k);
      v2f b3 = *(const v2f*)(yb3 + k);
      a0c0 = __builtin_amdgcn_wmma_f32_16x16x4_f32(false, a0, false, b0, (short)0, a0c0, false, false);
      a1c0 = __builtin_amdgcn_wmma_f32_16x16x4_f32(false, a1, false, b0, (short)0, a1c0, false, false);
      a0c1 = __builtin_amdgcn_wmma_f32_16x16x4_f32(false, a0, false, b1, (short)0, a0c1, false, false);
      a1c1 = __builtin_amdgcn_wmma_f32_16x16x4_f32(false, a1, false, b1, (short)0, a1c1, false, false);
      a0c2 = __builtin_amdgcn_wmma_f32_16x16x4_f32(false, a0, false, b2, (short)0, a0c2, false, false);
      a1c2 = __builtin_amdgcn_wmma_f32_16x16x4_f32(false, a1, false, b2, (short)0, a1c2, false, false);
      a0c3 = __builtin_amdgcn_wmma_f32_16x16x4_f32(false, a0, false, b3, (short)0, a0c3, false, false);
      a1c3 = __builtin_amdgcn_wmma_f32_16x16x4_f32(false, a1, false, b3, (short)0, a1c3, false, false);
    }

    float yv0 = (c0 < N) ? y2[c0] : INFINITY;
    float yv1 = (c1 < N) ? y2[c1] : INFINITY;
    float yv2 = (c2 < N) ? y2[c2] : INFINITY;
    float yv3 = (c3 < N) ? y2[c3] : INFINITY;
    #pragma unroll
    for (int r = 0; r < 8; ++r) {
      float s00 = yv0 - 2.f * a0c0[r], s10 = yv0 - 2.f * a1c0[r];
      float s01 = yv1 - 2.f * a0c1[r], s11 = yv1 - 2.f * a1c1[r];
      float s02 = yv2 - 2.f * a0c2[r], s12 = yv2 - 2.f * a1c2[r];
      float s03 = yv3 - 2.f * a0c3[r], s13 = yv3 - 2.f * a1c3[r];
      if (s00 < minv0[r]) { minv0[r] = s00; mini0[r] = c0; }
      if (s01 < minv0[r]) { minv0[r] = s01; mini0[r] = c1; }
      if (s02 < minv0[r]) { minv0[r] = s02; mini0[r] = c2; }
      if (s03 < minv0[r]) { minv0[r] = s03; mini0[r] = c3; }
      if (s10 < minv1[r]) { minv1[r] = s10; mini1[r] = c0; }
      if (s11 < minv1[r]) { minv1[r] = s11; mini1[r] = c1; }
      if (s12 < minv1[r]) { minv1[r] = s12; mini1[r] = c2; }
      if (s13 < minv1[r]) { minv1[r] = s13; mini1[r] = c3; }
    }
  }

  // tile memory is dead: alias LDS for the 32-row x 128-slot reduction
  __syncthreads();
  float* redV = smem;                      // 32*128 floats = 16 KB
  int*   redI = (int*)(smem + 32 * 128);   // 32*128 ints   = 16 KB

  #pragma unroll
  for (int r = 0; r < 8; ++r) {
    int m0 = r + 8 * hi;          // rows 0..15
    int m1 = m0 + 16;             // rows 16..31
    int slot = wave * 16 + lr;
    redV[m0 * 128 + slot] = minv0[r];  redI[m0 * 128 + slot] = mini0[r];
    redV[m1 * 128 + slot] = minv1[r];  redI[m1 * 128 + slot] = mini1[r];
  }
  __syncthreads();

  if (tid < 32 && qm + tid < Q) {
    float bv = INFINITY; int bi = 0x7fffffff;
    for (int s = 0; s < 128; ++s) {
      float v = redV[tid * 128 + s]; int idx = redI[tid * 128 + s];
      if (v < bv || (v == bv && idx < bi)) { bv = v; bi = idx; }
    }
    float d2 = x2[qm + tid] + bv;
    pscore[qm + tid] = sqrtf(fmaxf(d2, 0.f));
    ploc[qm + tid]   = bi;
  }
}

// ---------------------------------------------------------------------------
// Kernel 3: per-image argmax over 784 patch scores
// ---------------------------------------------------------------------------
__global__ __launch_bounds__(256) void pc_img_argmax(const float* __restrict__ pscore,
                                                     const int* __restrict__ ploc,
                                                     float* __restrict__ img_score,
                                                     int* __restrict__ img_patch,
                                                     int* __restrict__ img_loc, int P) {
  __shared__ float sv[256]; __shared__ int si[256];
  int b = blockIdx.x, tid = threadIdx.x;
  float bv = -INFINITY; int bi = 0x7fffffff;
  for (int p = tid; p < P; p += 256) {
    float v = pscore[b * P + p];
    if (v > bv || (v == bv && p < bi)) { bv = v; bi = p; }
  }
  sv[tid] = bv; si[tid] = bi; __syncthreads();
  for (int s = 128; s > 0; s >>= 1) {
    if (tid < s) {
      if (sv[tid + s] > sv[tid] || (sv[tid + s] == sv[tid] && si[tid + s] < si[tid])) {
        sv[tid] = sv[tid + s]; si[tid] = si[tid + s];
      }
    }
    __syncthreads();
  }
  if (tid == 0) {
    img_score[b] = sv[0];
    img_patch[b] = si[0];
    img_loc[b]   = ploc[b * P + si[0]];
  }
}

// ---------------------------------------------------------------------------
// Kernel 4: top-9 nearest bank rows to nn_sample (one block per image)
// ---------------------------------------------------------------------------
__global__ __launch_bounds__(256) void pc_topk(const float* __restrict__ Y,
                                               const int* __restrict__ img_loc,
                                               int* __restrict__ support, int N) {
  __shared__ float nn[DIMS];
  __shared__ float candV[8 * KNN_K]; __shared__ int candI[8 * KNN_K];
  int b = blockIdx.x, tid = threadIdx.x;
  int nnidx = img_loc[b];
  for (int k = tid; k < DIMS; k += 256) nn[k] = Y[(size_t)nnidx * DIMS + k];
  __syncthreads();

  int wave = tid >> 5, lane = tid & 31;
  float topV[KNN_K]; int topI[KNN_K];
  #pragma unroll
  for (int i = 0; i < KNN_K; ++i) { topV[i] = INFINITY; topI[i] = -1; }

  for (int j = wave; j < N; j += 8) {
    const float* yr = Y + (size_t)j * DIMS;
    float s = 0.f;
    #pragma unroll
    for (int k = lane; k < DIMS; k += 32) { float d = nn[k] - yr[k]; s += d * d; }
    #pragma unroll
    for (int o = 16; o > 0; o >>= 1) s += __shfl_xor(s, o, 32);
    if (lane == 0 && s < topV[KNN_K - 1]) {
      int i = KNN_K - 1;
      while (i > 0 && topV[i - 1] > s) { topV[i] = topV[i - 1]; topI[i] = topI[i - 1]; --i; }
      topV[i] = s; topI[i] = j;
    }
  }
  if (lane == 0)
    for (int i = 0; i < KNN_K; ++i) { candV[wave * KNN_K + i] = topV[i]; candI[wave * KNN_K + i] = topI[i]; }
  __syncthreads();

  if (tid == 0) {
    for (int sel = 0; sel < KNN_K; ++sel) {
      float bv = INFINITY; int bi = 0x7fffffff, bp = -1;
      for (int t = 0; t < 8 * KNN_K; ++t) {
        if (candI[t] < 0) continue;
        if (candV[t] < bv || (candV[t] == bv && candI[t] < bi)) { bv = candV[t]; bi = candI[t]; bp = t; }
      }
      support[b * KNN_K + sel] = (bp >= 0) ? bi : 0;
      if (bp >= 0) candI[bp] = -1;
    }
  }
}

// ---------------------------------------------------------------------------
// Kernel 5: softmax reweighting -> pred_scores   (one block/image, 9 waves)
// ---------------------------------------------------------------------------
__global__ __launch_bounds__(288) void pc_score(const float* __restrict__ X,
                                                const float* __restrict__ Y,
                                                const int* __restrict__ support,
                                                const float* __restrict__ img_score,
                                                const int* __restrict__ img_patch,
                                                float* __restrict__ out, int P) {
  __shared__ float dists[KNN_K];
  int b = blockIdx.x, tid = threadIdx.x;
  int wave = tid >> 5, lane = tid & 31;
  const float* mf = X + ((size_t)b * P + img_patch[b]) * DIMS;
  if (wave < KNN_K) {
    const float* yr = Y + (size_t)support[b * KNN_K + wave] * DIMS;
    float s = 0.f;
    #pragma unroll
    for (int k = lane; k < DIMS; k += 32) { float d = mf[k] - yr[k]; s += d * d; }
    #pragma unroll
    for (int o = 16; o > 0; o >>= 1) s += __shfl_xor(s, o, 32);
    if (lane == 0) dists[wave] = sqrtf(fmaxf(s, 0.f));
  }
  __syncthreads();
  if (tid == 0) {
    float mx = -INFINITY;
    for (int i = 0; i < KNN_K; ++i) mx = fmaxf(mx, dists[i]);
    float se = 0.f, e0 = 0.f;
    for (int i = 0; i < KNN_K; ++i) {
      float e = expf(dists[i] - mx);
      se += e; if (i == 0) e0 = e;
    }
    out[b] = (1.f - e0 / se) * img_score[b];
  }
}

// ---------------------------------------------------------------------------
// Kernel 6: bilinear 28x28 -> 224x224 (half-pixel centers, edge clamp)
// ---------------------------------------------------------------------------
__global__ __launch_bounds__(256) void pc_resize(const float* __restrict__ pscore,
                                                 float* __restrict__ amap, int total) {
  int idx = blockIdx.x * 256 + threadIdx.x;
  if (idx >= total) return;
  int b = idx / (OUT_HW * OUT_HW);
  int rem = idx - b * OUT_HW * OUT_HW;
  int oy = rem / OUT_HW, ox = rem - oy * OUT_HW;
  float sy = (oy + 0.5f) * ((float)IN_HW / OUT_HW) - 0.5f;
  float sx = (ox + 0.5f) * ((float)IN_HW / OUT_HW) - 0.5f;
  int y0 = (int)floorf(sy); float fy = sy - y0;
  int x0 = (int)floorf(sx); float fx = sx - x0;
  int y0c = min(max(y0, 0), IN_HW - 1), y1c = min(max(y0 + 1, 0), IN_HW - 1);
  int x0c = min(max(x0, 0), IN_HW - 1), x1c = min(max(x0 + 1, 0), IN_HW - 1);
  const float* p = pscore + b * IN_HW * IN_HW;
  float v00 = p[y0c * IN_HW + x0c], v01 = p[y0c * IN_HW + x1c];
  float v10 = p[y1c * IN_HW + x0c], v11 = p[y1c * IN_HW + x1c];
  amap[idx] = v00 * (1.f - fy) * (1.f - fx) + v01 * (1.f - fy) * fx
            + v10 * fy * (1.f - fx)        + v11 * fy * fx;
}

// ---------------------------------------------------------------------------
// Kernel 7: separable gaussian blur (sigma=4, 33 taps, zero padding as in
// conv 'SAME').  dir=0: along H; dir=1: along W.
// ---------------------------------------------------------------------------
__global__ __launch_bounds__(256) void pc_blur(const float* __restrict__ in,
                                               float* __restrict__ out,
                                               int total, int dir) {
  __shared__ float kc[2 * GRAD + 1];
  int tid = threadIdx.x;
  if (tid < 2 * GRAD + 1) { float x = (float)(tid - GRAD); kc[tid] = expf(-(x * x) / (2.f * 16.f)); }
  __syncthreads();
  if (tid == 0) {
    float s = 0.f;
    for (int i = 0; i < 2 * GRAD + 1; ++i) s += kc[i];
    for (int i = 0; i < 2 * GRAD + 1; ++i) kc[i] /= s;
  }
  __syncthreads();

  int idx = blockIdx.x * 256 + tid;
  if (idx >= total) return;
  int b = idx / (OUT_HW * OUT_HW);
  int rem = idx - b * OUT_HW * OUT_HW;
  int y = rem / OUT_HW, x = rem - y * OUT_HW;
  const float* base = in + (size_t)b * OUT_HW * OUT_HW;
  float s = 0.f;
  if (dir == 0) {
    #pragma unroll
    for (int t = -GRAD; t <= GRAD; ++t) {
      int yy = y + t;
      if (yy >= 0 && yy < OUT_HW) s += kc[t + GRAD] * base[yy * OUT_HW + x];
    }
  } else {
    #pragma unroll
    for (int t = -GRAD; t <= GRAD; ++t) {
      int xx = x + t;
      if (xx >= 0 && xx < OUT_HW) s += kc[t + GRAD] * base[y * OUT_HW + xx];
    }
  }
  out[idx] = s;
}

// ---------------------------------------------------------------------------
extern "C" void kernel_launch(void* const* d_in, const int* in_sizes, int n_in,
                              void* d_out, int out_size, void* d_ws, size_t ws_size,
                              hipStream_t stream) {
  const float* X = (const float*)d_in[0];    // embedding [Q, 384]
  const float* Y = (const float*)d_in[1];    // memory_bank [N, 384]
  const int Q = in_sizes[0] / DIMS;          // 6272
  const int N = in_sizes[1] / DIMS;          // 50000
  const int P = IN_HW * IN_HW;               // 784
  const int B = Q / P;                       // 8
  float* out = (float*)d_out;

  // scratch arena (256B aligned slices)
  char* ws = (char*)d_ws;
  size_t off = 0;
  auto arena = [&](size_t bytes) -> void* {
    void* p = ws + off;
    off = (off + bytes + 255) & ~(size_t)255;
    return p;
  };
  float* y2        = (float*)arena((size_t)N * 4);
  float* x2        = (float*)arena((size_t)Q * 4);
  float* pscore    = (float*)arena((size_t)Q * 4);
  int*   ploc      = (int*)  arena((size_t)Q * 4);
  float* img_score = (float*)arena((size_t)B * 4);
  int*   img_patch = (int*)  arena((size_t)B * 4);
  int*   img_loc   = (int*)  arena((size_t)B * 4);
  int*   support   = (int*)  arena((size_t)B * KNN_K * 4);
  const int maptotal = B * OUT_HW * OUT_HW;
  float* amap      = (float*)arena((size_t)maptotal * 4);
  float* tmpm      = (float*)arena((size_t)maptotal * 4);

  // 1) norms
  pc_row_norm<<<(Q + 7) / 8, 256, 0, stream>>>(X, x2, Q);
  pc_row_norm<<<(N + 7) / 8, 256, 0, stream>>>(Y, y2, N);

  // 2) fused WMMA distance GEMM + min/argmin (32 query rows per block)
  pc_knn_wmma<<<(Q + 31) / 32, 256, 0, stream>>>(X, Y, x2, y2, pscore, ploc, Q, N);

  // 3) per-image max patch
  pc_img_argmax<<<B, 256, 0, stream>>>(pscore, ploc, img_score, img_patch, img_loc, P);

  // 4) top-9 support set around each image's NN sample
  pc_topk<<<B, 256, 0, stream>>>(Y, img_loc, support, N);

  // 5) softmax-reweighted pred scores -> tail of d_out
  pc_score<<<B, 288, 0, stream>>>(X, Y, support, img_score, img_patch,
                                  out + (size_t)maptotal, P);

  // 6) bilinear upsample 28 -> 224
  pc_resize<<<(maptotal + 255) / 256, 256, 0, stream>>>(pscore, amap, maptotal);

  // 7) separable gaussian blur (H then W, matching reference conv order)
  pc_blur<<<(maptotal + 255) / 256, 256, 0, stream>>>(amap, tmpm, maptotal, 0);
  pc_blur<<<(maptotal + 255) / 256, 256, 0, stream>>>(tmpm, out, maptotal, 1);
}